// inner_62302795596402
// MI455X (gfx1250) — compile-verified
//
#include <hip/hip_runtime.h>
#include <hip/hip_bf16.h>

typedef __attribute__((ext_vector_type(2))) float v2f;
typedef __attribute__((ext_vector_type(8))) float v8f;

#define NCLS 16
#define DD 64
#define BLOCKS_PER_CLASS 16

// ---------------- utility kernels ----------------

__global__ void zero_kernel(float* p, int n) {
    int i = blockIdx.x * blockDim.x + threadIdx.x;
    if (i < n) p[i] = 0.0f;
}

__global__ void count_kernel(const long long* __restrict__ label, int* counts, int N) {
    int i = blockIdx.x * blockDim.x + threadIdx.x;
    if (i < N) atomicAdd(&counts[(int)label[i]], 1);
}

// class list offsets padded to 8-int alignment so per-chunk list reads are b64
__global__ void offset_kernel(const int* __restrict__ counts, int* offs, int* cursors) {
    if (threadIdx.x == 0 && blockIdx.x == 0) {
        int s = 0;
        for (int c = 0; c < NCLS; ++c) {
            offs[c] = s; cursors[c] = s;
            s += counts[c];
            s = (s + 7) & ~7;
        }
    }
}

__global__ void scatter_kernel(const long long* __restrict__ label, int* cursors,
                               int* __restrict__ list, int N) {
    int i = blockIdx.x * blockDim.x + threadIdx.x;
    if (i < N) {
        int c = (int)label[i];
        int pos = atomicAdd(&cursors[c], 1);
        list[pos] = i;
    }
}

// ---------------- stage 1: per-class Gram via fp32 WMMA ----------------

__device__ __forceinline__ void wmma16(v8f acc[4][4], const v2f frag[4]) {
#pragma unroll
    for (int ti = 0; ti < 4; ++ti)
#pragma unroll
        for (int tj = 0; tj < 4; ++tj)
            acc[ti][tj] = __builtin_amdgcn_wmma_f32_16x16x4_f32(
                false, frag[ti], false, frag[tj],
                (short)0, acc[ti][tj], false, false);
}

// Branchless fragment load for a complete 4-row chunk: one b64 list read +
// 8 dword feat reads per lane, no exec manipulation.
__device__ __forceinline__ void load_full(const float* __restrict__ feat,
                                          const int* __restrict__ list,
                                          int off, int ch, int m, int pair,
                                          v2f frag[4]) {
    const int2* lp = reinterpret_cast<const int2*>(list + off + ch * 4);
    int2 rr = lp[pair];
    const float* pA = feat + (long long)rr.x * DD + m;
    const float* pB = feat + (long long)rr.y * DD + m;
#pragma unroll
    for (int cg = 0; cg < 4; ++cg) {
        frag[cg][0] = pA[cg * 16];
        frag[cg][1] = pB[cg * 16];
    }
}

__global__ void __launch_bounds__(256) gram_kernel(const float* __restrict__ feat,
                                                   const int* __restrict__ list,
                                                   const int* __restrict__ counts,
                                                   const int* __restrict__ offs,
                                                   float* __restrict__ G) {
    int c    = blockIdx.x / BLOCKS_PER_CLASS;
    int blk  = blockIdx.x % BLOCKS_PER_CLASS;
    int lane = threadIdx.x & 31;
    int wave = threadIdx.x >> 5;
    int wavesPerBlock = blockDim.x >> 5;
    int waveId = blk * wavesPerBlock + wave;
    int nWaves = BLOCKS_PER_CLASS * wavesPerBlock;

    int cnt = counts[c];
    int off = offs[c];
    int nFull  = cnt >> 2;                       // complete 4-row chunks
    int tail   = cnt & 3;
    int nChunk = nFull + (tail ? 1 : 0);

    // contiguous chunk span per wave
    int per   = (nChunk + nWaves - 1) / nWaves;
    int start = waveId * per;
    int end   = min(start + per, nChunk);
    if (start >= end) return;

    int m    = lane & 15;   // column-within-16 / M / N index
    int pair = lane >> 4;   // 0: K rows {0,1}, 1: K rows {2,3}

    v8f acc[4][4];
#pragma unroll
    for (int ti = 0; ti < 4; ++ti)
#pragma unroll
        for (int tj = 0; tj < 4; ++tj) {
            v8f z = {};
            acc[ti][tj] = z;
        }

    int fullEnd = min(end, nFull);
    int ch = start;
    // 2x unrolled branchless hot loop: 18 loads issued together, then 32 WMMAs
    for (; ch + 1 < fullEnd; ch += 2) {
        v2f f0[4], f1[4];
        load_full(feat, list, off, ch,     m, pair, f0);
        load_full(feat, list, off, ch + 1, m, pair, f1);
        wmma16(acc, f0);
        wmma16(acc, f1);
    }
    for (; ch < fullEnd; ++ch) {
        v2f f0[4];
        load_full(feat, list, off, ch, m, pair, f0);
        wmma16(acc, f0);
    }
    // tail chunk (at most one per class, owned by one wave): clamped + selected
    if (tail && nFull >= start && nFull < end) {
        int r0 = nFull * 4 + pair * 2;
        int i0 = off + min(r0,     cnt - 1);
        int i1 = off + min(r0 + 1, cnt - 1);
        long long iA = list[i0];
        long long iB = list[i1];
        bool vA = r0 < cnt;
        bool vB = (r0 + 1) < cnt;
        const float* pA = feat + iA * DD + m;
        const float* pB = feat + iB * DD + m;
        v2f f0[4];
#pragma unroll
        for (int cg = 0; cg < 4; ++cg) {
            float a = pA[cg * 16];
            float b = pB[cg * 16];
            f0[cg][0] = vA ? a : 0.0f;
            f0[cg][1] = vB ? b : 0.0f;
        }
        wmma16(acc, f0);
    }

    float* Gc = G + c * DD * DD;
#pragma unroll
    for (int ti = 0; ti < 4; ++ti)
#pragma unroll
        for (int tj = 0; tj < 4; ++tj)
#pragma unroll
            for (int e = 0; e < 8; ++e) {
                int row = ti * 16 + e + pair * 8;   // C/D layout: vgpr e, lane-half -> M
                int col = tj * 16 + m;              // N = lane&15
                atomicAdd(&Gc[row * DD + col], acc[ti][tj][e]);
            }
}

// ---------------- stage 2: top eigenvector via repeated squaring ----------------
__global__ void __launch_bounds__(256) eig_kernel(const float* __restrict__ G,
                                                  float* __restrict__ V) {
    __shared__ float A[64][65];   // pad to dodge bank conflicts
    __shared__ float red[256];
    int c = blockIdx.x;
    int tid = threadIdx.x;
    const float* Gc = G + c * DD * DD;

    for (int t = tid; t < 4096; t += 256) A[t >> 6][t & 63] = Gc[t];
    __syncthreads();

    // normalize by max-abs
    float lm = 0.0f;
    for (int t = tid; t < 4096; t += 256) lm = fmaxf(lm, fabsf(A[t >> 6][t & 63]));
    red[tid] = lm; __syncthreads();
    for (int st = 128; st > 0; st >>= 1) {
        if (tid < st) red[tid] = fmaxf(red[tid], red[tid + st]);
        __syncthreads();
    }
    float inv0 = red[0] > 0.0f ? 1.0f / red[0] : 0.0f;
    __syncthreads();
    for (int t = tid; t < 4096; t += 256) A[t >> 6][t & 63] *= inv0;
    __syncthreads();

    int r  = tid >> 2;
    int cb = (tid & 3) * 16;
    for (int it = 0; it < 18; ++it) {          // A <- (A*A)/max|.|, ratio^(2^18)
        float s[16];
#pragma unroll
        for (int j = 0; j < 16; ++j) s[j] = 0.0f;
        for (int k = 0; k < 64; ++k) {
            float a = A[r][k];
#pragma unroll
            for (int j = 0; j < 16; ++j) s[j] = fmaf(a, A[k][cb + j], s[j]);
        }
        float loc = 0.0f;
#pragma unroll
        for (int j = 0; j < 16; ++j) loc = fmaxf(loc, fabsf(s[j]));
        red[tid] = loc; __syncthreads();
        for (int st = 128; st > 0; st >>= 1) {
            if (tid < st) red[tid] = fmaxf(red[tid], red[tid + st]);
            __syncthreads();
        }
        float inv = red[0] > 0.0f ? 1.0f / red[0] : 0.0f;
        __syncthreads();
#pragma unroll
        for (int j = 0; j < 16; ++j) A[r][cb + j] = s[j] * inv;
        __syncthreads();
    }

    if (tid == 0) {
        // A ~ lambda*v*v^T: column at largest diagonal = v; sign convention is
        // automatic (diag entry at argmax|v| is lambda*v^2 > 0)
        int jm = 0; float dm = A[0][0];
        for (int j = 1; j < 64; ++j) if (A[j][j] > dm) { dm = A[j][j]; jm = j; }
        float nrm = 0.0f;
        for (int j = 0; j < 64; ++j) nrm += A[j][jm] * A[j][jm];
        nrm = sqrtf(nrm);
        float inv = nrm > 0.0f ? 1.0f / nrm : 0.0f;
        for (int j = 0; j < 64; ++j) V[c * 64 + j] = A[j][jm] * inv;
        if (nrm == 0.0f) V[c * 64] = 1.0f;   // degenerate (empty class) fallback
    }
}

// ---------------- stage 3: out[i] = <feat_i, V[label_i]> ----------------
__global__ void __launch_bounds__(256) out_kernel(const float* __restrict__ feat,
                                                  const long long* __restrict__ label,
                                                  const float* __restrict__ V,
                                                  float* __restrict__ out, int N) {
    int lane = threadIdx.x & 31;
    int gwave = (blockIdx.x * blockDim.x + threadIdx.x) >> 5;
    int nWaves = (gridDim.x * blockDim.x) >> 5;
    int half = lane >> 4;
    int g = lane & 15;
    int nPairs = N >> 1;
    for (int p = gwave; p < nPairs; p += nWaves) {
        int row = p * 2 + half;
        const float4* f4 = reinterpret_cast<const float4*>(feat + (size_t)row * DD);
        float4 x = f4[g];
        int cls = (int)label[row];
        const float4* v4 = reinterpret_cast<const float4*>(V + cls * DD);
        float4 y = v4[g];
        float s = x.x * y.x + x.y * y.y + x.z * y.z + x.w * y.w;
        s += __shfl_xor(s, 1, 32);
        s += __shfl_xor(s, 2, 32);
        s += __shfl_xor(s, 4, 32);
        s += __shfl_xor(s, 8, 32);
        if (g == 0) out[row] = s;
    }
}

// ---------------- launcher ----------------
extern "C" void kernel_launch(void* const* d_in, const int* in_sizes, int n_in,
                              void* d_out, int out_size, void* d_ws, size_t ws_size,
                              hipStream_t stream) {
    const float* feat      = (const float*)d_in[0];
    const long long* label = (const long long*)d_in[1];
    int N = in_sizes[1];                         // 524288 rows

    // ws layout (dwords): G[16*64*64] | V[16*64] | counts[16] | offs[16] | cursors[16]
    //                     | list[N + 128]  (128 = per-class 8-int alignment padding)
    float* G      = (float*)d_ws;
    float* V      = G + NCLS * DD * DD;
    int*   counts = (int*)(V + NCLS * DD);
    int*   offs    = counts + NCLS;
    int*   cursors = offs + NCLS;
    int*   list    = cursors + NCLS;
    float* out = (float*)d_out;

    // zero G, V, counts (contiguous); offs/cursors overwritten; list fully written
    int zn = NCLS * DD * DD + NCLS * DD + NCLS;
    zero_kernel<<<(zn + 255) / 256, 256, 0, stream>>>(G, zn);

    count_kernel<<<(N + 255) / 256, 256, 0, stream>>>(label, counts, N);
    offset_kernel<<<1, 32, 0, stream>>>(counts, offs, cursors);
    scatter_kernel<<<(N + 255) / 256, 256, 0, stream>>>(label, cursors, list, N);

    gram_kernel<<<NCLS * BLOCKS_PER_CLASS, 256, 0, stream>>>(feat, list, counts, offs, G);
    eig_kernel<<<NCLS, 256, 0, stream>>>(G, V);
    out_kernel<<<2048, 256, 0, stream>>>(feat, label, V, out, N);
}